// WordTokenRepresentation_20624432956205
// MI455X (gfx1250) — compile-verified
//
#include <hip/hip_runtime.h>
#include <hip/hip_bf16.h>
#include <stdint.h>

// Problem constants (match reference file).
#define B_   16
#define L_   2048
#define H_   1024
#define W_   512
#define TPW  4        // tokens per word
#define G_   8        // words processed per block (double-buffered)
#define NT   256      // threads per block (8 wave32)
#define WCOLS 128     // float columns per wave (32 lanes x float4)

typedef unsigned int v4u __attribute__((ext_vector_type(4)));
typedef int          v8i __attribute__((ext_vector_type(8)));
typedef int          v4i __attribute__((ext_vector_type(4)));

// ---------------------------------------------------------------------------
// Issue one TDM descriptor per wave: move the 4 token rows x 128 float
// columns slice of word `w` owned by wave `wv` into LDS buffer `buf`.
// Tile lands contiguously in LDS: row r at (slice_base + r*512B).
// Tracked by TENSORcnt (+1 per call, in-order per wave).
// ---------------------------------------------------------------------------
__device__ __forceinline__ void tdm_load_slice(const float* __restrict__ hs,
                                               int b, int w, int buf,
                                               unsigned lds_base, int wv) {
  // Global byte address of this wave's tile start.
  const unsigned long long gaddr =
      (unsigned long long)(uintptr_t)hs +
      (((unsigned long long)(b * L_ + w * TPW) * H_ + (unsigned)(wv * WCOLS)) * 4ull);
  // LDS byte address of this wave's slice in buffer `buf`.
  const unsigned lds_addr =
      lds_base + (unsigned)(buf * TPW * H_ * 4) + (unsigned)(wv * TPW * WCOLS * 4);

  // D# group 0: count=1 (valid user descriptor), lds_addr, global_addr, type=2.
  v4u g0;
  g0.x = 1u;                                    // [1:0] count=1, user mode
  g0.y = lds_addr;                              // [63:32] lds_addr (bytes)
  g0.z = (unsigned)(gaddr & 0xffffffffull);     // [95:64] global_addr[31:0]
  g0.w = (unsigned)((gaddr >> 32) & 0x01ffffffull)  // [120:96] global_addr[56:32]
       | (2u << 30);                            // [127:126] type = 2 ("image")

  // D# group 1: data_size=4B (code 2), 2D tensor:
  //   tensor_dim0 = 128, tensor_dim1 = 4, tile_dim0 = 128, tile_dim1 = 4,
  //   tensor_dim0_stride = H = 1024 elements.
  v8i g1;
  g1[0] = (int)(2u << 16);        // [15:0] wg_mask=0, [17:16] data_size=2 (4B)
  g1[1] = (int)(128u << 16);      // [63:48] tensor_dim0[15:0] = 128
  g1[2] = (int)(4u << 16);        // [95:80] tensor_dim1[15:0] = 4
  g1[3] = (int)(128u << 16);      // [127:112] tile_dim0 = 128
  g1[4] = 4;                      // [143:128] tile_dim1 = 4 (tile_dim2 = 0)
  g1[5] = 1024;                   // [191:160] tensor_dim0_stride = 1024
  g1[6] = 0;                      // stride0 hi / stride1 lo (unused, 2D)
  g1[7] = 0;

  const v4i g2 = {0, 0, 0, 0};    // dims 2/3 unused (2D tensor)
  const v4i g3 = {0, 0, 0, 0};    // dim 4 unused
  const v8i g4 = {0, 0, 0, 0, 0, 0, 0, 0};  // extra group (6-arg form), unused

  __builtin_amdgcn_tensor_load_to_lds(g0, g1, g2, g3, g4, 0);
}

// ---------------------------------------------------------------------------
// Pooling kernel: block = (batch b, group of G_ words). Wave wv owns float
// columns [wv*128, wv*128+128); lane owns a float4 of that span. Each wave
// TDM-stages its own slice (double buffered), so only s_wait_tensorcnt is
// needed -- no cross-wave barriers.
// ---------------------------------------------------------------------------
__global__ void __launch_bounds__(NT)
word_pool_kernel(const float* __restrict__ hs,
                 const int*   __restrict__ tmask,
                 const int*   __restrict__ widx,
                 float*       __restrict__ out) {
  const int tid  = threadIdx.x;
  const int lane = tid & 31;
  const int wv   = __builtin_amdgcn_readfirstlane(tid >> 5);  // wave id (SGPR)
  const int b    = blockIdx.y;
  const int w0   = blockIdx.x * G_;

  __shared__ float lds[2 * TPW * H_];           // 32 KB, double buffered
  const unsigned lds_base =
      (unsigned)__builtin_amdgcn_readfirstlane((int)(uintptr_t)(void*)lds);

  // Prologue: prefetch word w0 into buffer 0.
  tdm_load_slice(hs, b, w0, 0, lds_base, wv);

  for (int i = 0; i < G_; ++i) {
    const int w   = w0 + i;
    const int buf = i & 1;

    if (i + 1 < G_) {
      // Prefetch next word into the other buffer; TDM ops complete in order
      // per wave, so <=1 outstanding means word i's tile has landed.
      tdm_load_slice(hs, b, w + 1, buf ^ 1, lds_base, wv);
      __builtin_amdgcn_s_wait_tensorcnt(1);
    } else {
      __builtin_amdgcn_s_wait_tensorcnt(0);
    }
    asm volatile("" ::: "memory");  // order LDS reads after the wait

    // Token validity for this word (uniform across block; broadcast loads).
    const int tb = b * L_ + w * TPW;
    float sel[TPW];
    float cnt = 0.0f;
#pragma unroll
    for (int r = 0; r < TPW; ++r) {
      const int v = (tmask[tb + r] == 1) && (widx[tb + r] == w);
      sel[r] = v ? 1.0f : 0.0f;
      cnt += sel[r];
    }
    const float inv = 1.0f / fmaxf(cnt, 1.0f);

    // Branchless masked accumulate of the 4 token rows from this wave's
    // LDS slice: slice is TPW rows x 128 floats, row-major.
    const int slice = buf * TPW * H_ + wv * TPW * WCOLS;
    float4 acc = make_float4(0.f, 0.f, 0.f, 0.f);
#pragma unroll
    for (int r = 0; r < TPW; ++r) {
      const float4 v = *reinterpret_cast<const float4*>(
          &lds[slice + r * WCOLS + lane * 4]);
      acc.x += sel[r] * v.x;
      acc.y += sel[r] * v.y;
      acc.z += sel[r] * v.z;
      acc.w += sel[r] * v.w;
    }
    acc.x *= inv; acc.y *= inv; acc.z *= inv; acc.w *= inv;

    // Column = wv*128 + lane*4 == tid*4.
    *reinterpret_cast<float4*>(&out[((size_t)(b * W_ + w)) * H_ + tid * 4]) = acc;
  }
}

// ---------------------------------------------------------------------------
// word_lengths[b] = max over valid tokens of word_index + 1  (int64)
// ---------------------------------------------------------------------------
__global__ void __launch_bounds__(NT)
word_len_kernel(const int* __restrict__ tmask,
                const int* __restrict__ widx,
                long long* __restrict__ wlen) {
  const int b = blockIdx.x;
  int m = -1;
  for (int t = threadIdx.x; t < L_; t += NT) {
    const int idx = widx[b * L_ + t];
    const int v = (tmask[b * L_ + t] == 1) && (idx >= 0);
    if (v && idx > m) m = idx;
  }
  __shared__ int red[NT];
  red[threadIdx.x] = m;
  __syncthreads();
#pragma unroll
  for (int s = NT / 2; s > 0; s >>= 1) {
    if (threadIdx.x < s) {
      const int o = red[threadIdx.x + s];
      if (o > red[threadIdx.x]) red[threadIdx.x] = o;
    }
    __syncthreads();
  }
  if (threadIdx.x == 0) wlen[b] = (long long)(red[0] + 1);
}

// ---------------------------------------------------------------------------
extern "C" void kernel_launch(void* const* d_in, const int* in_sizes, int n_in,
                              void* d_out, int out_size, void* d_ws, size_t ws_size,
                              hipStream_t stream) {
  (void)in_sizes; (void)n_in; (void)out_size; (void)d_ws; (void)ws_size;

  const float* hs    = (const float*)d_in[0];   // [B, L, H] f32
  const int*   tmask = (const int*)  d_in[1];   // [B, L]
  const int*   widx  = (const int*)  d_in[2];   // [B, L]
  // d_in[3] = num_words scalar (compile-time W_ here)

  float* out = (float*)d_out;                                 // [B, W, H] f32
  long long* wlen = (long long*)(out + (size_t)B_ * W_ * H_); // [B] int64 tail

  dim3 grid(W_ / G_, B_);   // 64 x 16 blocks
  word_pool_kernel<<<grid, NT, 0, stream>>>(hs, tmask, widx, out);
  word_len_kernel<<<dim3(B_), NT, 0, stream>>>(tmask, widx, wlen);
}